// TorchScriptGNN_36378372997637
// MI455X (gfx1250) — compile-verified
//
#include <hip/hip_runtime.h>

typedef __attribute__((ext_vector_type(2))) float v2f;
typedef __attribute__((ext_vector_type(4))) float v4f;
typedef __attribute__((ext_vector_type(8))) float v8f;

#define NFEAT 128   // feature / hidden width (both 128 in this model)

// ---------------------------------------------------------------------------
// zero fill (float4 granularity; all buffers here are multiples of 4 floats)
// ---------------------------------------------------------------------------
__global__ __launch_bounds__(256) void k_zero4(float* __restrict__ p, int n4) {
  int i = blockIdx.x * blockDim.x + threadIdx.x;
  if (i < n4) ((v4f*)p)[i] = (v4f){0.f, 0.f, 0.f, 0.f};
}

// ---------------------------------------------------------------------------
// deg[row[e]] += 1   (hardware fp32 atomic, no return -> STOREcnt path)
// ---------------------------------------------------------------------------
__global__ __launch_bounds__(256) void k_degree(const int* __restrict__ row,
                                                float* __restrict__ deg, int E) {
  int i = blockIdx.x * blockDim.x + threadIdx.x;
  if (i < E) unsafeAtomicAdd(&deg[row[i]], 1.0f);
}

// dinv = deg > 0 ? rsqrt(deg) : 0
__global__ __launch_bounds__(256) void k_dinv(const float* __restrict__ deg,
                                              float* __restrict__ dinv, int n) {
  int i = blockIdx.x * blockDim.x + threadIdx.x;
  if (i < n) {
    float d = deg[i];
    dinv[i] = (d > 0.f) ? rsqrtf(d) : 0.f;
  }
}

// ---------------------------------------------------------------------------
// Edge aggregation: agg[row] += x[col] * (dinv[row]*dinv[col])
// One wave (32 lanes) per edge; each lane moves a float4 -> one full 128-float
// row per wave. Gathers hit L2 (feature table = 51.2 MB < 192 MB L2).
// ---------------------------------------------------------------------------
__global__ __launch_bounds__(256) void k_edge_agg(const float* __restrict__ x,
                                                  const int* __restrict__ row,
                                                  const int* __restrict__ col,
                                                  const float* __restrict__ dinv,
                                                  float* __restrict__ agg, int E) {
  long tid = (long)blockIdx.x * blockDim.x + threadIdx.x;
  int e = (int)(tid >> 5);
  if (e >= E) return;
  int lane = threadIdx.x & 31;
  int r = row[e];
  int c = col[e];
  float norm = dinv[r] * dinv[c];
  v4f xv = ((const v4f*)(x + (long)c * NFEAT))[lane];
  float* dst = agg + (long)r * NFEAT + lane * 4;
  unsafeAtomicAdd(dst + 0, xv.x * norm);
  unsafeAtomicAdd(dst + 1, xv.y * norm);
  unsafeAtomicAdd(dst + 2, xv.z * norm);
  unsafeAtomicAdd(dst + 3, xv.w * norm);
}

// ---------------------------------------------------------------------------
// out[n, j] = relu( sum_k A[n,k] * W[j,k] + bias[j] )   (A: [nrows,128],
// W: [128,128] row-major [out][in]).  fp32 WMMA 16x16x4, one 16x16 tile per
// wave, 8 waves per block cover all 128 output columns; K-loop of 32 WMMAs.
// Bias folded into accumulator init (each C VGPR of a lane holds column
// N = lane%16, so a scalar broadcast initializes the whole column).
// nrows must be a multiple of 16 (100000 = 6250*16).
// ---------------------------------------------------------------------------
__global__ __launch_bounds__(256) void k_gemm_bias_relu(const float* __restrict__ A,
                                                        const float* __restrict__ W,
                                                        const float* __restrict__ bias,
                                                        float* __restrict__ out,
                                                        int relu) {
  const int wave = threadIdx.x >> 5;      // 0..7 -> column tile
  const int lane = threadIdx.x & 31;
  const int half = lane >> 4;             // 0: K pair (k,k+1), 1: (k+2,k+3)
  const int lm   = lane & 15;             // M index for A, N index for B/C
  const int row0 = blockIdx.x * 16;
  const int col0 = wave * 16;

  const float* arow = A + (long)(row0 + lm) * NFEAT;  // A[M = lm][*]
  const float* wrow = W + (long)(col0 + lm) * NFEAT;  // B[k][N=lm] = W[col0+lm][k]

  float bv = bias[col0 + lm];
  v8f acc = {bv, bv, bv, bv, bv, bv, bv, bv};

#pragma unroll
  for (int k = 0; k < NFEAT; k += 4) {
    int keff = k + half * 2;
    v2f a = *(const v2f*)(arow + keff);   // A: VGPR0=K(keff), VGPR1=K(keff+1)
    v2f b = *(const v2f*)(wrow + keff);   // B: mirrored layout
    acc = __builtin_amdgcn_wmma_f32_16x16x4_f32(
        /*neg_a=*/false, a, /*neg_b=*/false, b,
        /*c_mod=*/(short)0, acc, /*reuse_a=*/false, /*reuse_b=*/false);
  }

  // C/D layout: VGPR r -> M = r + 8*half, N = lm
#pragma unroll
  for (int r = 0; r < 8; ++r) {
    float v = acc[r];
    if (relu) v = fmaxf(v, 0.f);
    out[(long)(row0 + r + 8 * half) * NFEAT + col0 + lm] = v;
  }
}

// ---------------------------------------------------------------------------
// pooled[batch[n]] += h[n]   (segment sum; one wave per node, float4 lanes)
// ---------------------------------------------------------------------------
__global__ __launch_bounds__(256) void k_pool(const float* __restrict__ h,
                                              const int* __restrict__ batch,
                                              float* __restrict__ pooled, int N) {
  long tid = (long)blockIdx.x * blockDim.x + threadIdx.x;
  int n = (int)(tid >> 5);
  if (n >= N) return;
  int lane = threadIdx.x & 31;
  int g = batch[n];
  v4f hv = ((const v4f*)(h + (long)n * NFEAT))[lane];
  float* dst = pooled + (long)g * NFEAT + lane * 4;
  unsafeAtomicAdd(dst + 0, hv.x);
  unsafeAtomicAdd(dst + 1, hv.y);
  unsafeAtomicAdd(dst + 2, hv.z);
  unsafeAtomicAdd(dst + 3, hv.w);
}

// ---------------------------------------------------------------------------
// head: out[g,c] = pooled[g] . w_lin[c] + b_lin[c]   (128x2 outputs, tiny)
// ---------------------------------------------------------------------------
__global__ __launch_bounds__(256) void k_head(const float* __restrict__ pooled,
                                              const float* __restrict__ wlin,
                                              const float* __restrict__ blin,
                                              float* __restrict__ out) {
  int t = threadIdx.x;          // 0..255 ; g = t/2, c = t%2
  int g = t >> 1;
  int c = t & 1;
  const float* pr = pooled + g * NFEAT;
  const float* wr = wlin + c * NFEAT;
  float s = blin[c];
#pragma unroll
  for (int k = 0; k < NFEAT; k += 4) {
    v4f a = *(const v4f*)(pr + k);
    v4f b = *(const v4f*)(wr + k);
    s += a.x * b.x + a.y * b.y + a.z * b.z + a.w * b.w;
  }
  out[t] = s;
}

// ---------------------------------------------------------------------------
extern "C" void kernel_launch(void* const* d_in, const int* in_sizes, int n_in,
                              void* d_out, int out_size, void* d_ws, size_t ws_size,
                              hipStream_t stream) {
  const float* x     = (const float*)d_in[0];
  const int*   ei    = (const int*)d_in[1];
  const int*   batch = (const int*)d_in[2];
  const float* w1    = (const float*)d_in[3];
  const float* b1    = (const float*)d_in[4];
  const float* w2    = (const float*)d_in[5];
  const float* b2    = (const float*)d_in[6];
  const float* wlin  = (const float*)d_in[7];
  const float* blin  = (const float*)d_in[8];

  const int E = in_sizes[1] / 2;     // edge_index is [2, E]
  const int N = in_sizes[2];         // batch has one entry per node
  const int* row = ei;
  const int* col = ei + E;

  // workspace layout (fp32): deg[N] | dinv[N] | agg[N*128] | h[N*128] | pooled[128*128]
  float* deg    = (float*)d_ws;
  float* dinv   = deg + N;
  float* agg    = dinv + N;
  float* h      = agg + (size_t)N * NFEAT;
  float* pooled = h + (size_t)N * NFEAT;

  const dim3 B(256);
  const long featElems = (long)N * NFEAT;
  const int  zb_feat = (int)((featElems / 4 + 255) / 256);
  const int  zb_deg  = (N / 4 + 255) / 256;
  const int  eg_blocks = (int)(((long)E * 32 + 255) / 256);
  const int  pl_blocks = (int)(((long)N * 32 + 255) / 256);
  const int  gemm_blocks = N / 16;   // N = 100000 -> exact

  // degree + normalization
  k_zero4<<<zb_deg, B, 0, stream>>>(deg, N / 4);
  k_degree<<<(E + 255) / 256, B, 0, stream>>>(row, deg, E);
  k_dinv<<<(N + 255) / 256, B, 0, stream>>>(deg, dinv, N);

  // layer 1
  k_zero4<<<zb_feat, B, 0, stream>>>(agg, (int)(featElems / 4));
  k_edge_agg<<<eg_blocks, B, 0, stream>>>(x, row, col, dinv, agg, E);
  k_gemm_bias_relu<<<gemm_blocks, B, 0, stream>>>(agg, w1, b1, h, 1);

  // layer 2
  k_zero4<<<zb_feat, B, 0, stream>>>(agg, (int)(featElems / 4));
  k_edge_agg<<<eg_blocks, B, 0, stream>>>(h, row, col, dinv, agg, E);
  k_gemm_bias_relu<<<gemm_blocks, B, 0, stream>>>(agg, w2, b2, h, 1);

  // pooling + head
  k_zero4<<<(128 * 128 / 4 + 255) / 256, B, 0, stream>>>(pooled, 128 * 128 / 4);
  k_pool<<<pl_blocks, B, 0, stream>>>(h, batch, pooled, N);
  k_head<<<1, B, 0, stream>>>(pooled, wlin, blin, (float*)d_out);
}